// Model_35347580846430
// MI455X (gfx1250) — compile-verified
//
#include <hip/hip_runtime.h>

// CDNA5 / gfx1250, wave32. Depthwise moving average (W=25, reflect pad 12)
// expressed as banded-matrix x slab matmul on V_WMMA_F32_16X16X4_F32.

typedef float v2f __attribute__((ext_vector_type(2)));
typedef float v8f __attribute__((ext_vector_type(8)));

#define T_DIM 4096
#define N_DIM 128
#define WIN   25
#define PAD   12
#define TT    64                 // time steps per workgroup
#define SLAB  (TT + 2 * PAD)     // 88 slab rows in LDS
#define LSTR  136                // LDS row stride (floats): 2*136 % 64 == 16 -> upper half-wave
                                 // hits a disjoint 16-bank group (conflict-free ds_load_b32)
#define NTILE (TT / 16)          // 4 time tiles per wave
#define KCH   10                 // K = 40 = 10 chunks of 4

__global__ __launch_bounds__(256) void trend_ma25_wmma(const float* __restrict__ x,
                                                       float* __restrict__ out) {
  __shared__ float lds[SLAB * LSTR];

  const int tid      = threadIdx.x;
  const int bid      = blockIdx.x;
  const int tblocks  = T_DIM / TT;           // 64
  const int b        = bid / tblocks;
  const int t0       = (bid % tblocks) * TT;

  // ---------------- stage slab rows [t0-PAD, t0+TT+PAD) into LDS (reflect at batch edges)
  const float* xb = x + (size_t)b * T_DIM * N_DIM;
  for (int f = tid; f < SLAB * N_DIM; f += 256) {   // 88*128/256 = 44 exact iterations
    int row = f >> 7;            // /128
    int col = f & (N_DIM - 1);
    int t = t0 - PAD + row;
    t = (t < 0) ? -t : t;                          // reflect low
    t = (t >= T_DIM) ? (2 * T_DIM - 2 - t) : t;    // reflect high
    lds[row * LSTR + col] = xb[(size_t)t * N_DIM + col];
  }
  __syncthreads();

  const int lane  = tid & 31;
  const int wave  = tid >> 5;
  const int i     = lane & 15;             // M (A/C rows) or N (B/C cols) within the tile
  const int khalf = (lane >> 4) << 1;      // lanes 16..31 carry K+2 (f32 A/B VGPR layout)
  const int nb    = wave * 16;             // channel base for this wave

  // ---------------- constant banded coefficient matrix as A fragments:
  // C[i,j] = (i <= j <= i+24) ? 1/25 : 0, j in [0,40)
  v2f a[KCH];
#pragma unroll
  for (int c = 0; c < KCH; ++c) {
    const int j0 = 4 * c + khalf;
    a[c].x = (j0     >= i && j0     <= i + (WIN - 1)) ? (1.0f / WIN) : 0.0f;
    a[c].y = (j0 + 1 >= i && j0 + 1 <= i + (WIN - 1)) ? (1.0f / WIN) : 0.0f;
  }

#pragma unroll
  for (int tt = 0; tt < NTILE; ++tt) {
    v8f acc = {0.f, 0.f, 0.f, 0.f, 0.f, 0.f, 0.f, 0.f};
#pragma unroll
    for (int c = 0; c < KCH; ++c) {
      const int rs = 16 * tt + 4 * c + khalf;      // slab row of this K pair
      v2f bfrag;
      bfrag.x = lds[(rs    ) * LSTR + nb + i];     // B[K=4c+khalf  , N=i]
      bfrag.y = lds[(rs + 1) * LSTR + nb + i];     // B[K=4c+khalf+1, N=i]
      // D = A*B + C  (8 args: neg_a, A, neg_b, B, c_mod, C, reuse_a, reuse_b)
      acc = __builtin_amdgcn_wmma_f32_16x16x4_f32(false, a[c], false, bfrag,
                                                  (short)0, acc, false, false);
    }
    // ---------------- store D tile: lane holds (M = r + 8*hi, N = i), r = VGPR 0..7
    const int mhalf = (lane >> 4) << 3;
    float* op = out + ((size_t)b * T_DIM + (size_t)(t0 + 16 * tt + mhalf)) * N_DIM + nb + i;
#pragma unroll
    for (int r = 0; r < 8; ++r) {
      op[(size_t)r * N_DIM] = acc[r];
    }
  }
}

extern "C" void kernel_launch(void* const* d_in, const int* in_sizes, int n_in,
                              void* d_out, int out_size, void* d_ws, size_t ws_size,
                              hipStream_t stream) {
  (void)n_in; (void)out_size; (void)d_ws; (void)ws_size;
  const float* x = (const float*)d_in[0];
  float* out     = (float*)d_out;
  const int B    = in_sizes[0] / (T_DIM * N_DIM);   // 64
  const int grid = B * (T_DIM / TT);                // 64 * 64 = 4096 workgroups of 8 waves
  trend_ma25_wmma<<<grid, 256, 0, stream>>>(x, out);
}